// one_batch_net_29850022707280
// MI455X (gfx1250) — compile-verified
//
#include <hip/hip_runtime.h>
#include <cstdint>
#include <cstddef>

typedef float v2f __attribute__((ext_vector_type(2)));
typedef float v8f __attribute__((ext_vector_type(8)));

// CDNA5 f32 matrix core: D(16x16,f32) = A(16x4,f32) x B(4x16,f32) + C
__device__ __forceinline__ v8f wmma_f32x4(v2f a, v2f b, v8f c) {
  return __builtin_amdgcn_wmma_f32_16x16x4_f32(
      /*neg_a=*/false, a, /*neg_b=*/false, b,
      /*c_mod=*/(short)0, c, /*reuse_a=*/false, /*reuse_b=*/false);
}

// CDNA5 hardware tanh (v_tanh_f32, TRANS32) when available; ocml fallback.
__device__ __forceinline__ float fast_tanh(float x) {
#if __has_builtin(__builtin_amdgcn_tanhf)
  return __builtin_amdgcn_tanhf(x);
#elif __has_builtin(__builtin_amdgcn_tanh_f32)
  return __builtin_amdgcn_tanh_f32(x);
#else
  return tanhf(x);
#endif
}

constexpr int kB = 8, kN = 192, kS = 128, kT = 2;
constexpr int kF0 = 25, kF1 = 50, kF2 = 100, kAX = 16;
constexpr int kHID = 240;
constexpr int kD = kF2 * kAX;          // 1600
constexpr int kAtoms = kB * kN;        // 1536
// K-padded LDS row strides (multiples of 4 for 16x16x4 k-steps)
constexpr int sG0 = 28, sG1 = 52, sG2 = 112;
// padded filter-weight tiles staged in LDS
constexpr int sW1K = 28, sW1N = 64;    // fw1: 25x50 -> 28x64
constexpr int sW2K = 52, sW2N = 112;   // fw2: 50x100 -> 52x112

// ---------------------------------------------------------------------------
// Kernel 1: per (atom, type) -> descriptor D (100x16) into workspace.
// 4 waves / 128 threads. LDS regions overlapped by liveness:
//   region1 [0,6656):        G1 (128x52)  then RG (16x112)
//   region2 [6656,20992):    G0 (128x28)  then G2 (128x112)
//   region3 [20992,23040):   Rt (16x128, rows 4..15 zero)
//   region4 [23040,24832):   Wf1 (28x64, zero-padded fw1)
//   region5 [24832,30656):   Wf2 (52x112, zero-padded fw2)
// ---------------------------------------------------------------------------
__global__ __launch_bounds__(128) void k_descriptor(
    const float* __restrict__ sym,
    const float* __restrict__ fw0, const float* __restrict__ fb0,
    const float* __restrict__ fw1, const float* __restrict__ fb1,
    const float* __restrict__ fw2, const float* __restrict__ fb2,
    float* __restrict__ wsD)
{
  __shared__ float smem[30656];
  float* G1  = smem;           // 128 x sG1
  float* RG  = smem;           // 16 x sG2 (overlays dead G1)
  float* G0  = smem + 6656;    // 128 x sG0
  float* G2  = smem + 6656;    // 128 x sG2 (overlays dead G0)
  float* Rt  = smem + 20992;   // 16 x 128
  float* Wf1 = smem + 23040;   // 28 x 64
  float* Wf2 = smem + 24832;   // 52 x 112

  const int atom = blockIdx.x;
  const int t    = blockIdx.y;
  const int tid  = threadIdx.x;
  const int lane = tid & 31;
  const int wave = tid >> 5;
  const int lrow = lane & 15;
  const int lhi  = lane >> 4;

  // stage zero-padded weights (once per block; guards live only here)
  for (int idx = tid; idx < sW1K * sW1N; idx += 128) {
    const int k = idx >> 6, n = idx & 63;
    Wf1[idx] = (k < kF0 && n < kF1) ? fw1[((size_t)t * kF0 + k) * kF1 + n] : 0.f;
  }
  for (int idx = tid; idx < sW2K * sW2N; idx += 128) {
    const int k = idx / sW2N, n = idx % sW2N;
    Wf2[idx] = (k < kF1 && n < kF2) ? fw2[((size_t)t * kF1 + k) * kF2 + n] : 0.f;
  }

  // layer 0: G0[s][f] = tanh(s * fw0 + fb0), 1 -> 25 (zero-pad to 28);
  // also stage zero-padded R^T (16 x 128) so RG's A-frags are plain LDS loads.
  {
    const float4 r4 = ((const float4*)sym)[(size_t)atom * kS + tid];
#pragma unroll
    for (int r = 0; r < 16; ++r)
      Rt[r * kS + tid] = (r == 0) ? r4.x : (r == 1) ? r4.y
                       : (r == 2) ? r4.z : (r == 3) ? r4.w : 0.f;
    const float* w0 = fw0 + t * kF0;
    const float* b0 = fb0 + t * kF0;
    for (int f = 0; f < sG0; ++f)
      G0[tid * sG0 + f] = (f < kF0) ? fast_tanh(r4.x * w0[f] + b0[f]) : 0.f;
  }
  __syncthreads();

  // layer 1: G1 = tanh(G0 @ fw1 + fb1), (128x25)@(25x50), N padded to 64
  for (int mt = wave; mt < 8; mt += 4) {
    v2f a[7];
#pragma unroll
    for (int ks = 0; ks < 7; ++ks)
#pragma unroll
      for (int i = 0; i < 2; ++i)
        a[ks][i] = G0[(mt * 16 + lrow) * sG0 + ks * 4 + i + 2 * lhi];
    for (int nt = 0; nt < 4; ++nt) {
      const int n = nt * 16 + lrow;
      v8f acc = {};
#pragma unroll
      for (int ks = 0; ks < 7; ++ks) {
        v2f b;
#pragma unroll
        for (int i = 0; i < 2; ++i)
          b[i] = Wf1[(ks * 4 + i + 2 * lhi) * sW1N + n];
        acc = wmma_f32x4(a[ks], b, acc);
      }
      if (n < sG1) {
#pragma unroll
        for (int r = 0; r < 8; ++r) {
          const int row = mt * 16 + r + 8 * lhi;
          G1[row * sG1 + n] = (n < kF1) ? fast_tanh(acc[r] + fb1[t * kF1 + n]) : 0.f;
        }
      }
    }
  }
  __syncthreads();

  // layer 2: G2 = tanh(G1 @ fw2 + fb2), (128x50)@(50x100), N padded to 112
  for (int mt = wave; mt < 8; mt += 4) {
    v2f a[13];
#pragma unroll
    for (int ks = 0; ks < 13; ++ks)
#pragma unroll
      for (int i = 0; i < 2; ++i)
        a[ks][i] = G1[(mt * 16 + lrow) * sG1 + ks * 4 + i + 2 * lhi];
    for (int nt = 0; nt < 7; ++nt) {
      const int n = nt * 16 + lrow;
      v8f acc = {};
#pragma unroll
      for (int ks = 0; ks < 13; ++ks) {
        v2f b;
#pragma unroll
        for (int i = 0; i < 2; ++i)
          b[i] = Wf2[(ks * 4 + i + 2 * lhi) * sW2N + n];
        acc = wmma_f32x4(a[ks], b, acc);
      }
#pragma unroll
      for (int r = 0; r < 8; ++r) {
        const int row = mt * 16 + r + 8 * lhi;
        G2[row * sG2 + n] = (n < kF2) ? fast_tanh(acc[r] + fb2[t * kF2 + n]) : 0.f;
      }
    }
  }
  __syncthreads();

  // RG = R^T @ G2 : (16x128)@(128x112); each wave owns n-tiles {wave, wave+4}
  {
    const int nt0 = wave;
    const int nt1 = wave + 4;
    const bool has1 = (nt1 < 7);           // wave-uniform
    v8f acc0 = {}, acc1 = {};
#pragma unroll 4
    for (int ks = 0; ks < 32; ++ks) {
      v2f a, b0, b1;
#pragma unroll
      for (int i = 0; i < 2; ++i) {
        const int s = ks * 4 + i + 2 * lhi;
        a[i]  = Rt[lrow * kS + s];
        b0[i] = G2[s * sG2 + nt0 * 16 + lrow];
        b1[i] = has1 ? G2[s * sG2 + nt1 * 16 + lrow] : 0.f;
      }
      acc0 = wmma_f32x4(a, b0, acc0);
      acc1 = wmma_f32x4(a, b1, acc1);
    }
#pragma unroll
    for (int r = 0; r < 8; ++r)
      RG[(r + 8 * lhi) * sG2 + nt0 * 16 + lrow] = acc0[r];
    if (has1) {
#pragma unroll
      for (int r = 0; r < 8; ++r)
        RG[(r + 8 * lhi) * sG2 + nt1 * 16 + lrow] = acc1[r];
    }
  }
  __syncthreads();

  // D[m][a] = sum_f RG[f][m] * RG[f][a], K = 4 (one WMMA k-step per m-tile)
  for (int mt = wave; mt < 7; mt += 4) {
    v2f a, b;
#pragma unroll
    for (int i = 0; i < 2; ++i) {
      const int f = i + 2 * lhi;
      a[i] = RG[f * sG2 + mt * 16 + lrow];
      b[i] = RG[f * sG2 + lrow];
    }
    v8f acc = {};
    acc = wmma_f32x4(a, b, acc);
#pragma unroll
    for (int r = 0; r < 8; ++r) {
      const int m = mt * 16 + r + 8 * lhi;
      if (m < kF2)
        wsD[((size_t)(t * kAtoms + atom)) * kD + m * kAX + lrow] = acc[r];
    }
  }
}

// ---------------------------------------------------------------------------
// Kernel 2: fitting net over tiles of 16 atoms x type.
// 160 threads = 5 waves x 3 N-tiles = 240 cols exactly.
// ---------------------------------------------------------------------------
__device__ __forceinline__ void fit_layer240(
    const float* __restrict__ src, float* __restrict__ dst,
    const float* __restrict__ w, const float* __restrict__ bias,
    int lrow, int lhi, int nbase)
{
  v8f acc[3] = {{}, {}, {}};
#pragma unroll 4
  for (int ks = 0; ks < 60; ++ks) {
    v2f a;
#pragma unroll
    for (int i = 0; i < 2; ++i)
      a[i] = src[lrow * kHID + ks * 4 + i + 2 * lhi];
#pragma unroll
    for (int j = 0; j < 3; ++j) {
      v2f b;
#pragma unroll
      for (int i = 0; i < 2; ++i) {
        const int k = ks * 4 + i + 2 * lhi;
        b[i] = w[(size_t)k * kHID + nbase + j * 16 + lrow];
      }
      acc[j] = wmma_f32x4(a, b, acc[j]);
    }
  }
#pragma unroll
  for (int j = 0; j < 3; ++j) {
#pragma unroll
    for (int r = 0; r < 8; ++r) {
      const int row = r + 8 * lhi;
      const int col = nbase + j * 16 + lrow;
      dst[row * kHID + col] = fast_tanh(acc[j][r] + bias[col]);
    }
  }
}

__global__ __launch_bounds__(160) void k_fitting(
    const float* __restrict__ wsD, const int* __restrict__ types,
    const float* __restrict__ gw0, const float* __restrict__ gb0,
    const float* __restrict__ gw1, const float* __restrict__ gb1,
    const float* __restrict__ gw2, const float* __restrict__ gb2,
    const float* __restrict__ gow, const float* __restrict__ gob,
    float* __restrict__ out)
{
  __shared__ float h0[16 * kHID];
  __shared__ float h1[16 * kHID];
  __shared__ float Dch[16 * 64];

  const int atile = blockIdx.x;
  const int t     = blockIdx.y;
  const int tid   = threadIdx.x;
  const int lane  = tid & 31;
  const int wave  = tid >> 5;
  const int lrow  = lane & 15;
  const int lhi   = lane >> 4;
  const int nbase = wave * 48;   // this wave's 3 N-tiles

  // layer g0: h0 = tanh(D(16x1600) @ gw0 + gb0), K streamed via LDS chunks
  v8f acc[3] = {{}, {}, {}};
  const float* w0 = gw0 + (size_t)t * kD * kHID;
  for (int kc = 0; kc < kD; kc += 64) {
    __syncthreads();
    for (int idx = tid; idx < 16 * 64; idx += 160) {
      const int row = idx >> 6, k = idx & 63;
      Dch[idx] = wsD[((size_t)(t * kAtoms + atile * 16 + row)) * kD + kc + k];
    }
    __syncthreads();
#pragma unroll 4
    for (int ks = 0; ks < 16; ++ks) {
      v2f a;
#pragma unroll
      for (int i = 0; i < 2; ++i)
        a[i] = Dch[lrow * 64 + ks * 4 + i + 2 * lhi];
#pragma unroll
      for (int j = 0; j < 3; ++j) {
        v2f b;
#pragma unroll
        for (int i = 0; i < 2; ++i) {
          const int k = kc + ks * 4 + i + 2 * lhi;
          b[i] = w0[(size_t)k * kHID + nbase + j * 16 + lrow];
        }
        acc[j] = wmma_f32x4(a, b, acc[j]);
      }
    }
  }
  {
    const float* b0 = gb0 + t * kHID;
#pragma unroll
    for (int j = 0; j < 3; ++j) {
#pragma unroll
      for (int r = 0; r < 8; ++r) {
        const int row = r + 8 * lhi;
        const int col = nbase + j * 16 + lrow;
        h0[row * kHID + col] = fast_tanh(acc[j][r] + b0[col]);
      }
    }
  }
  __syncthreads();

  fit_layer240(h0, h1, gw1 + (size_t)t * kHID * kHID, gb1 + t * kHID, lrow, lhi, nbase);
  __syncthreads();
  fit_layer240(h1, h0, gw2 + (size_t)t * kHID * kHID, gb2 + t * kHID, lrow, lhi, nbase);
  __syncthreads();

  // e = h @ gow + gob, gather by type, accumulate per-batch sums
  if (tid < 16) {
    const int atom = atile * 16 + tid;
    if (types[atom] == t) {
      const float* gw = gow + t * kHID;
      float e = gob[t];
      for (int n = 0; n < kHID; ++n) e += h0[tid * kHID + n] * gw[n];
      atomicAdd(&out[atom / kN], e);
    }
  }
}

__global__ void k_zero_out(float* __restrict__ out) {
  if (threadIdx.x < kB) out[threadIdx.x] = 0.f;
}

// ---------------------------------------------------------------------------
extern "C" void kernel_launch(void* const* d_in, const int* in_sizes, int n_in,
                              void* d_out, int out_size, void* d_ws, size_t ws_size,
                              hipStream_t stream) {
  const float* sym  = (const float*)d_in[0];
  const int*   typ  = (const int*)  d_in[1];
  const float* fw0  = (const float*)d_in[2];
  const float* fb0  = (const float*)d_in[3];
  const float* fw1  = (const float*)d_in[4];
  const float* fb1  = (const float*)d_in[5];
  const float* fw2  = (const float*)d_in[6];
  const float* fb2  = (const float*)d_in[7];
  const float* gw0  = (const float*)d_in[8];
  const float* gb0  = (const float*)d_in[9];
  const float* gw1  = (const float*)d_in[10];
  const float* gb1  = (const float*)d_in[11];
  const float* gw2  = (const float*)d_in[12];
  const float* gb2  = (const float*)d_in[13];
  const float* gow  = (const float*)d_in[14];
  const float* gob  = (const float*)d_in[15];
  float* out = (float*)d_out;
  float* wsD = (float*)d_ws;   // T * 1536 * 1600 f32 = 19.7 MB

  k_zero_out<<<1, 32, 0, stream>>>(out);
  k_descriptor<<<dim3(kAtoms, kT), 128, 0, stream>>>(
      sym, fw0, fb0, fw1, fb1, fw2, fb2, wsD);
  k_fitting<<<dim3(kAtoms / 16, kT), 160, 0, stream>>>(
      wsD, typ, gw0, gb0, gw1, gb1, gw2, gb2, gow, gob, out);
}